// TorchBaseConv2D3x3_5093831213684
// MI455X (gfx1250) — compile-verified
//
#include <hip/hip_runtime.h>

// ---------------------------------------------------------------------------
// Sparse 3x3 conv (gather-GEMM) + bias + BatchNorm for MI455X (gfx1250).
// fp32 WMMA (V_WMMA_F32_16X16X4_F32), software-pipelined double-buffered LDS.
// ---------------------------------------------------------------------------

typedef __attribute__((ext_vector_type(2))) float v2f;
typedef __attribute__((ext_vector_type(4))) float v4f;
typedef __attribute__((ext_vector_type(8))) float v8f;

#define N_TOT   262144            // B * N_PER_BATCH
#define C       64                // C_IN == C_OUT
#define MTILE   128               // rows per block
#define NBLKS   (N_TOT / MTILE)   // 2048
#define LDA     68                // padded stride -> conflict-free A b64 loads
#define LDW     68                // padded stride -> conflict-free B b64 loads
#define ASZ     (MTILE * LDA)     // floats per A buffer
#define WSZ     (C * LDW)         // floats per W buffer
#define EPS_BN  1e-3f

// ---------------------------------------------------------------------------
// Kernel 1: pipelined gather + 9-tap GEMM + bias, per-block channel partials.
// ---------------------------------------------------------------------------
__global__ __launch_bounds__(256) void spconv_gemm(
    const float* __restrict__ feat,     // [N][64]
    const float* __restrict__ weight,   // [9][64][64]
    const float* __restrict__ bias,     // [64]
    const int*   __restrict__ nbr,      // [N][9]
    float*       __restrict__ out,      // [N][64]
    float*       __restrict__ psum,     // [NBLKS][64]
    float*       __restrict__ psq)      // [NBLKS][64]
{
    __shared__ float s_a[2 * ASZ];      // double-buffered gathered A tiles
    __shared__ float s_w[2 * WSZ];      // double-buffered transposed weights
    __shared__ int   s_idx[MTILE * 9];  // neighbor indices for this block

    const int t    = threadIdx.x;
    const int lane = t & 31;
    const int wave = t >> 5;
    const int col  = lane & 15;         // N/M index within half-wave
    const int hi   = lane >> 4;         // K-pair selector (16x16x4 layout)
    const int m0   = wave * 16;         // this wave's row sub-tile
    const int rowbase = blockIdx.x * MTILE;
    const int q    = t & 15;            // float4 column within a row
    const int gr   = t >> 4;            // row-group (0..15)
    const int cout = t & 63;
    const int cin0 = t >> 6;            // 0..3

    for (int e = t; e < MTILE * 9; e += 256)
        s_idx[e] = nbr[rowbase * 9 + e];

    // Accumulators: wave covers 16 rows x 64 cols = 4 f32 16x16 tiles.
    v8f acc[4];
#pragma unroll
    for (int nt = 0; nt < 4; ++nt) {
        const float bv = bias[nt * 16 + col];
#pragma unroll
        for (int j = 0; j < 8; ++j) acc[nt][j] = bv;
    }

    const float4* feat4 = (const float4*)feat;   // row i -> feat4[i*16 + q]
    float4 rf[8];    // register staging: gathered features for next tap
    float  rw[16];   // register staging: weights for next tap

    __syncthreads();   // s_idx ready

    // Issue global loads for tap k into registers (no wait forced here:
    // invalid-row masking is deferred to store time).
    auto prefetch = [&](int k) {
        const float* wk = weight + k * C * C;
#pragma unroll
        for (int p = 0; p < 8; ++p) {
            const int row = p * 16 + gr;
            const int idx = s_idx[row * 9 + k];
            const int src = (idx < 0) ? 0 : idx;
            rf[p] = feat4[src * 16 + q];                    // global_load_b128
        }
#pragma unroll
        for (int p = 0; p < 16; ++p)
            rw[p] = wk[(p * 4 + cin0) * C + cout];          // coalesced b32
    };

    // Drain staged registers into LDS buffer `buf` (waits on loads here).
    auto stage = [&](int k, int buf) {
        float* a = s_a + buf * ASZ;
        float* w = s_w + buf * WSZ;
#pragma unroll
        for (int p = 0; p < 8; ++p) {
            const int row = p * 16 + gr;
            float4 v = rf[p];
            if (s_idx[row * 9 + k] < 0) v = make_float4(0.f, 0.f, 0.f, 0.f);
            *(float4*)&a[row * LDA + q * 4] = v;            // ds_store_b128
        }
#pragma unroll
        for (int p = 0; p < 16; ++p)
            w[cout * LDW + p * 4 + cin0] = rw[p];           // transposed W
    };

    // 16x64 += A(16x64) * W(64x64) via 64 x V_WMMA_F32_16X16X4_F32.
    auto gemm = [&](int buf) {
        const float* a = s_a + buf * ASZ;
        const float* w = s_w + buf * WSZ;
#pragma unroll
        for (int kk = 0; kk < C; kk += 4) {
            const v2f av = *(const v2f*)&a[(m0 + col) * LDA + kk + 2 * hi];
#pragma unroll
            for (int nt = 0; nt < 4; ++nt) {
                const v2f bv = *(const v2f*)&w[(nt * 16 + col) * LDW + kk + 2 * hi];
                acc[nt] = __builtin_amdgcn_wmma_f32_16x16x4_f32(
                    false, av, false, bv, (short)0, acc[nt], false, false);
            }
        }
    };

    // -------- software pipeline: one barrier per tap --------
    prefetch(0);
    stage(0, 0);
    __syncthreads();
    for (int k = 0; k < 9; ++k) {
        if (k < 8) prefetch(k + 1);       // loads for k+1 fly during gemm(k)
        gemm(k & 1);
        if (k < 8) {
            stage(k + 1, (k + 1) & 1);    // safe: gemm(k-1) fenced by barrier
            __syncthreads();
        }
    }

    // -------- epilogue: LDS transpose -> coalesced stores + reduction ------
    __syncthreads();                       // all waves done reading buffers
#pragma unroll
    for (int nt = 0; nt < 4; ++nt)
#pragma unroll
        for (int jj = 0; jj < 8; ++jj)     // C layout: M = jj + 8*hi
            s_a[(m0 + jj + 8 * hi) * LDA + nt * 16 + col] = acc[nt][jj];
    __syncthreads();

    // Pre-BN output: fully coalesced global_store_b128.
#pragma unroll
    for (int p = 0; p < 8; ++p) {
        const int row = p * 16 + gr;
        const float4 v = *(const float4*)&s_a[row * LDA + q * 4];
        *(float4*)&out[(size_t)(rowbase + row) * C + q * 4] = v;
    }

    // Deterministic per-channel sum / sumsq over this block's 128 rows.
    float* s_red  = s_w;                   // [4][64]
    float* s_red2 = s_w + 4 * C;           // [4][64]
    {
        const int c = t & 63, quad = t >> 6;
        float s = 0.f, qq = 0.f;
#pragma unroll
        for (int r = 0; r < 32; ++r) {
            const float v = s_a[(quad * 32 + r) * LDA + c];
            s += v;  qq += v * v;
        }
        s_red [quad * C + c] = s;
        s_red2[quad * C + c] = qq;
    }
    __syncthreads();
    if (t < C) {
        float s = 0.f, qq = 0.f;
#pragma unroll
        for (int r = 0; r < 4; ++r) { s += s_red[r * C + t]; qq += s_red2[r * C + t]; }
        psum[blockIdx.x * C + t] = s;
        psq [blockIdx.x * C + t] = qq;
    }
}

// ---------------------------------------------------------------------------
// Kernel 2: reduce partials -> per-channel scale/shift.
// ---------------------------------------------------------------------------
__global__ __launch_bounds__(256) void bn_stats(
    const float* __restrict__ psum, const float* __restrict__ psq,
    const float* __restrict__ gamma, const float* __restrict__ beta,
    float* __restrict__ ss)   // ss[0..63]=scale, ss[64..127]=shift
{
    __shared__ float r1[4 * C], r2[4 * C];
    const int c = threadIdx.x & 63, g = threadIdx.x >> 6;
    float s = 0.f, qv = 0.f;
    for (int b = g; b < NBLKS; b += 4) {
        s  += psum[b * C + c];
        qv += psq [b * C + c];
    }
    r1[g * C + c] = s;
    r2[g * C + c] = qv;
    __syncthreads();
    if (threadIdx.x < C) {
        float st = 0.f, qt = 0.f;
#pragma unroll
        for (int r = 0; r < 4; ++r) { st += r1[r * C + threadIdx.x]; qt += r2[r * C + threadIdx.x]; }
        const float inv_n = 1.0f / (float)N_TOT;
        const float mean  = st * inv_n;
        const float var   = qt * inv_n - mean * mean;
        const float scale = gamma[threadIdx.x] * rsqrtf(var + EPS_BN);
        ss[threadIdx.x]     = scale;
        ss[C + threadIdx.x] = beta[threadIdx.x] - mean * scale;
    }
}

// ---------------------------------------------------------------------------
// Kernel 3: streaming BN apply; nontemporal b128 stores (out_bn never reused,
// keep the L2-resident pre-BN array hot instead).
// ---------------------------------------------------------------------------
__global__ __launch_bounds__(256) void bn_apply(
    const float* __restrict__ out, const float* __restrict__ ss,
    float* __restrict__ outbn)
{
    const int i = blockIdx.x * 256 + threadIdx.x;   // one float4 per thread
    const v4f* o4 = (const v4f*)out;
    const v4f* s4 = (const v4f*)ss;
    v4f*       d4 = (v4f*)outbn;
    const int c4 = i & 15;
    const v4f sc = s4[c4];
    const v4f sh = s4[16 + c4];
    const v4f v  = o4[i] * sc + sh;
    __builtin_nontemporal_store(v, &d4[i]);
}

// ---------------------------------------------------------------------------
extern "C" void kernel_launch(void* const* d_in, const int* in_sizes, int n_in,
                              void* d_out, int out_size, void* d_ws, size_t ws_size,
                              hipStream_t stream) {
    const float* feat  = (const float*)d_in[0];
    const float* wgt   = (const float*)d_in[1];
    const float* bias  = (const float*)d_in[2];
    const float* gamma = (const float*)d_in[3];
    const float* beta  = (const float*)d_in[4];
    const int*   nbr   = (const int*)d_in[5];

    float* out   = (float*)d_out;                    // [N][64] pre-BN
    float* outbn = out + (size_t)N_TOT * C;          // [N][64] post-BN

    float* psum = (float*)d_ws;                      // [NBLKS][64]
    float* psq  = psum + (size_t)NBLKS * C;          // [NBLKS][64]
    float* ss   = psq  + (size_t)NBLKS * C;          // [128] scale|shift

    spconv_gemm<<<NBLKS, 256, 0, stream>>>(feat, wgt, bias, nbr, out, psum, psq);
    bn_stats<<<1, 256, 0, stream>>>(psum, psq, gamma, beta, ss);
    bn_apply<<<(N_TOT * C) / 4 / 256, 256, 0, stream>>>(out, ss, outbn);
}